// GraphConvolution_13675175871113
// MI455X (gfx1250) — compile-verified
//
#include <hip/hip_runtime.h>
#include <hip/hip_bf16.h>

typedef __attribute__((ext_vector_type(2))) float v2f;
typedef __attribute__((ext_vector_type(8))) float v8f;

#define D 128  // D_IN == D_OUT == 128

// -------------------------------------------------------------------------
// Kernel 1: support = X @ W  via V_WMMA_F32_16X16X4_F32.
// One wave32 per 16x16 output tile. blockIdx.x = M-tile, wave-in-block = N-tile.
// A(16x4 f32): lane l<16 holds row l, K = k+0/k+1 in 2 VGPRs;
//              lane l>=16 holds row l-16, K = k+2/k+3.
// B(4x16): vgpr r holds K=r (lanes 0-15) / K=r+2 (lanes 16-31), N = lane&15.
// C/D(16x16): vgpr r holds M = r + 8*(lane>>4), N = lane&15.
// -------------------------------------------------------------------------
__global__ void gcn_gemm_wmma(const float* __restrict__ X,
                              const float* __restrict__ W,
                              float* __restrict__ S) {
  const int lane  = threadIdx.x & 31;
  const int ntile = threadIdx.x >> 5;     // 0..7  (N = 128 -> 8 tiles)
  const int half  = lane >> 4;            // 0 or 1
  const int l15   = lane & 15;
  const int m0    = blockIdx.x * 16;
  const int n0    = ntile * 16;

  const float* arow = X + (size_t)(m0 + l15) * D + half * 2;
  const float* bcol = W + n0 + l15;

  v8f c = {};
  #pragma unroll 4
  for (int k = 0; k < D; k += 4) {
    v2f a = *(const v2f*)(arow + k);                 // K = k+2*half, k+2*half+1
    v2f b;
    b.x = bcol[(size_t)(k + half * 2 + 0) * D];
    b.y = bcol[(size_t)(k + half * 2 + 1) * D];
    c = __builtin_amdgcn_wmma_f32_16x16x4_f32(
            /*neg_a=*/false, a, /*neg_b=*/false, b,
            /*c_mod=*/(short)0, c, /*reuse_a=*/false, /*reuse_b=*/false);
  }

  float* srow = S + (size_t)(m0 + half * 8) * D + n0 + l15;
  #pragma unroll
  for (int r = 0; r < 8; ++r) srow[(size_t)r * D] = c[r];
}

// -------------------------------------------------------------------------
// Kernel 2: out[n][d] = bias[d]  (must run every call: atomics accumulate)
// -------------------------------------------------------------------------
__global__ void gcn_init_out(float* __restrict__ out,
                             const float* __restrict__ bias, int total) {
  int i = blockIdx.x * blockDim.x + threadIdx.x;
  if (i < total) out[i] = bias[i & (D - 1)];
}

// -------------------------------------------------------------------------
// Kernel 3: for each edge e: out[row[e]] += val[e] * support[col[e]]
// One wave32 per edge iteration; lane owns 4 contiguous floats (b128 gather,
// 4x global_atomic_add_f32). support & out are L2-resident (51MB each < 192MB).
// -------------------------------------------------------------------------
__global__ void gcn_edge_scatter(const float* __restrict__ S,
                                 const int* __restrict__ erow,
                                 const int* __restrict__ ecol,
                                 const float* __restrict__ eval,
                                 float* __restrict__ out, int nEdges) {
  const int lane   = threadIdx.x & 31;
  const int wave   = (blockIdx.x * blockDim.x + threadIdx.x) >> 5;
  const int nWaves = (gridDim.x * blockDim.x) >> 5;

  for (int e = wave; e < nEdges; e += nWaves) {
    const int   r = erow[e];
    const int   ccol = ecol[e];
    const float v = eval[e];

    const float4 m = *((const float4*)(S + (size_t)ccol * D) + lane);
    float* dst = out + (size_t)r * D + lane * 4;
    atomicAdd(dst + 0, v * m.x);
    atomicAdd(dst + 1, v * m.y);
    atomicAdd(dst + 2, v * m.z);
    atomicAdd(dst + 3, v * m.w);
  }
}

// -------------------------------------------------------------------------
extern "C" void kernel_launch(void* const* d_in, const int* in_sizes, int n_in,
                              void* d_out, int out_size, void* d_ws, size_t ws_size,
                              hipStream_t stream) {
  const float* x     = (const float*)d_in[0];
  const float* w     = (const float*)d_in[1];
  const float* bias  = (const float*)d_in[2];
  const int*   erow  = (const int*)d_in[3];
  const int*   ecol  = (const int*)d_in[4];
  const float* evals = (const float*)d_in[5];
  float*       out   = (float*)d_out;

  const int nNodes = in_sizes[0] / D;
  const int nEdges = in_sizes[3];

  float* support = (float*)d_ws;   // nNodes * D floats = 51.2 MB

  // 1) support = x @ W   (one wave per 16x16 tile; 8 N-tiles per 256-thr block)
  const int mTiles = nNodes / 16;  // 100000 / 16 = 6250 exactly
  gcn_gemm_wmma<<<mTiles, 256, 0, stream>>>(x, w, support);

  // 2) out = bias (broadcast)
  const int total = nNodes * D;
  gcn_init_out<<<(total + 255) / 256, 256, 0, stream>>>(out, bias, total);

  // 3) edge scatter-add (one wave per edge, grid-stride)
  gcn_edge_scatter<<<8192, 256, 0, stream>>>(support, erow, ecol, evals,
                                             out, nEdges);
}